// CAM_76476187672967
// MI455X (gfx1250) — compile-verified
//
#include <hip/hip_runtime.h>
#include <hip/hip_bf16.h>

typedef __attribute__((ext_vector_type(2))) float v2f;
typedef __attribute__((ext_vector_type(8))) float v8f;

// D = A(16x4 f32) * B(4x16 f32) + C(16x16 f32)
#define WMMA_F32(ACC, A, B) \
  __builtin_amdgcn_wmma_f32_16x16x4_f32(false, (A), false, (B), (short)0, (ACC), false, false)

// Select k-pair (x,y) or (z,w) from a float4 based on half-wave id.
__device__ __forceinline__ v2f selpair(float4 v, int hw) {
  v2f r;
  r.x = hw ? v.z : v.x;
  r.y = hw ? v.w : v.y;
  return r;
}

static constexpr int C = 512;
static constexpr int N = 4096;  // 64*64

// ---------------------------------------------------------------------------
// Stage 1: phi/theta/g = W @ xf   ([C,C] x [C,N] per batch, 3 weights)
// grid: (N/64, C/64, B*3), block: 128 (4 waves, each one 16-row o-tile, NT=4)
// ---------------------------------------------------------------------------
__global__ __launch_bounds__(128) void cam_proj(
    const float* __restrict__ x, const float* __restrict__ w_phi,
    const float* __restrict__ w_theta, const float* __restrict__ w_g,
    float* __restrict__ phi, float* __restrict__ theta, float* __restrict__ g) {
  const int z = blockIdx.z;
  const int b = z / 3, widx = z % 3;
  const float* W = (widx == 0) ? w_phi : (widx == 1) ? w_theta : w_g;
  float* O = (widx == 0) ? phi : (widx == 1) ? theta : g;

  const int wave = threadIdx.x >> 5;
  const int lane = threadIdx.x & 31;
  const int hw = lane >> 4;
  const int l15 = lane & 15;
  const int o0 = (blockIdx.y * 4 + wave) * 16;
  const int n0 = blockIdx.x * 64;

  const float* xb = x + (size_t)b * C * N;
  const float* arow = W + (size_t)(o0 + l15) * C;  // weight row, K contiguous
  v8f acc[4] = {};

  for (int k = 0; k < C; k += 4) {
    float4 a4 = *(const float4*)(arow + k);
    v2f a = selpair(a4, hw);
    const float* bp = xb + (size_t)(k + 2 * hw) * N + n0 + l15;
#pragma unroll
    for (int t = 0; t < 4; ++t) {
      v2f bv;
      bv.x = bp[16 * t];
      bv.y = bp[N + 16 * t];
      acc[t] = WMMA_F32(acc[t], a, bv);
    }
  }

  float* ob = O + (size_t)b * C * N;
#pragma unroll
  for (int r = 0; r < 8; ++r) {
    const int row = o0 + r + 8 * hw;
    float* p = ob + (size_t)row * N + n0 + l15;
#pragma unroll
    for (int t = 0; t < 4; ++t) p[16 * t] = acc[t][r];
  }
}

// ---------------------------------------------------------------------------
// Stage 2: attn_raw[b,c,d] = sum_n phi[b,c,n]*theta[b,d,n]  (K = N = 4096)
// Both operands row-major over K -> both use the per-lane float4 trick.
// grid: (C/64, C/64, B), block 128
// ---------------------------------------------------------------------------
__global__ __launch_bounds__(128) void cam_logits(
    const float* __restrict__ phi, const float* __restrict__ theta,
    float* __restrict__ attn) {
  const int b = blockIdx.z;
  const int wave = threadIdx.x >> 5;
  const int lane = threadIdx.x & 31;
  const int hw = lane >> 4;
  const int l15 = lane & 15;
  const int c0 = (blockIdx.y * 4 + wave) * 16;
  const int d0 = blockIdx.x * 64;

  const float* pa = phi + (size_t)b * C * N + (size_t)(c0 + l15) * N;
  const float* pb0 = theta + (size_t)b * C * N + (size_t)(d0 + l15) * N;
  v8f acc[4] = {};

  for (int k = 0; k < N; k += 4) {
    float4 a4 = *(const float4*)(pa + k);
    v2f a = selpair(a4, hw);
#pragma unroll
    for (int t = 0; t < 4; ++t) {
      float4 b4 = *(const float4*)(pb0 + (size_t)(16 * t) * N + k);
      v2f bv = selpair(b4, hw);
      acc[t] = WMMA_F32(acc[t], a, bv);
    }
  }

  float* ab = attn + (size_t)b * C * C;
#pragma unroll
  for (int r = 0; r < 8; ++r) {
    const int row = c0 + r + 8 * hw;
    float* p = ab + (size_t)row * C + d0 + l15;
#pragma unroll
    for (int t = 0; t < 4; ++t) p[16 * t] = acc[t][r];
  }
}

// ---------------------------------------------------------------------------
// Stage 3: softmax over last dim of attn [B*C rows of 512], in place
// ---------------------------------------------------------------------------
__global__ __launch_bounds__(128) void cam_softmax(float* __restrict__ attn) {
  __shared__ float red[128];
  float* p = attn + (size_t)blockIdx.x * C;
  const int t = threadIdx.x;
  float v0 = p[t], v1 = p[t + 128], v2 = p[t + 256], v3 = p[t + 384];

  float m = fmaxf(fmaxf(v0, v1), fmaxf(v2, v3));
  red[t] = m;
  __syncthreads();
  for (int s = 64; s > 0; s >>= 1) {
    if (t < s) red[t] = fmaxf(red[t], red[t + s]);
    __syncthreads();
  }
  m = red[0];
  __syncthreads();

  float e0 = __expf(v0 - m), e1 = __expf(v1 - m);
  float e2 = __expf(v2 - m), e3 = __expf(v3 - m);
  red[t] = e0 + e1 + e2 + e3;
  __syncthreads();
  for (int s = 64; s > 0; s >>= 1) {
    if (t < s) red[t] += red[t + s];
    __syncthreads();
  }
  const float inv = 1.0f / red[0];
  p[t] = e0 * inv;
  p[t + 128] = e1 * inv;
  p[t + 256] = e2 * inv;
  p[t + 384] = e3 * inv;
}

// ---------------------------------------------------------------------------
// Stage 4: y = attn @ g, stored directly in the torch-permuted layout:
//   yt[b, n>>3, ((n&7)<<9) + c] = y[b, c, n]   (since N = 8*C)
// grid: (N/64, C/64, B), block 128
// ---------------------------------------------------------------------------
__global__ __launch_bounds__(128) void cam_apply(
    const float* __restrict__ attn, const float* __restrict__ g,
    float* __restrict__ yt) {
  const int b = blockIdx.z;
  const int wave = threadIdx.x >> 5;
  const int lane = threadIdx.x & 31;
  const int hw = lane >> 4;
  const int l15 = lane & 15;
  const int c0 = (blockIdx.y * 4 + wave) * 16;
  const int n0 = blockIdx.x * 64;

  const float* pa = attn + (size_t)b * C * C + (size_t)(c0 + l15) * C;
  const float* gb = g + (size_t)b * C * N;
  v8f acc[4] = {};

  for (int k = 0; k < C; k += 4) {
    float4 a4 = *(const float4*)(pa + k);
    v2f a = selpair(a4, hw);
    const float* bp = gb + (size_t)(k + 2 * hw) * N + n0 + l15;
#pragma unroll
    for (int t = 0; t < 4; ++t) {
      v2f bv;
      bv.x = bp[16 * t];
      bv.y = bp[N + 16 * t];
      acc[t] = WMMA_F32(acc[t], a, bv);
    }
  }

  float* ytb = yt + (size_t)b * C * N;
#pragma unroll
  for (int r = 0; r < 8; ++r) {
    const int c = c0 + r + 8 * hw;
#pragma unroll
    for (int t = 0; t < 4; ++t) {
      const int n = n0 + 16 * t + l15;
      ytb[(size_t)(n >> 3) * N + ((n & 7) << 9) + c] = acc[t][r];
    }
  }
}

// ---------------------------------------------------------------------------
// Stage 5: out = gamma * (w_mask @ yt) + x   (clean coalesced GEMM + epilogue)
// grid: (N/64, C/64, B), block 128
// ---------------------------------------------------------------------------
__global__ __launch_bounds__(128) void cam_mask(
    const float* __restrict__ yt, const float* __restrict__ w_mask,
    const float* __restrict__ x, const float* __restrict__ gamma,
    float* __restrict__ out) {
  const int b = blockIdx.z;
  const int wave = threadIdx.x >> 5;
  const int lane = threadIdx.x & 31;
  const int hw = lane >> 4;
  const int l15 = lane & 15;
  const int o0 = (blockIdx.y * 4 + wave) * 16;
  const int m0 = blockIdx.x * 64;

  const float* arow = w_mask + (size_t)(o0 + l15) * C;
  const float* yb = yt + (size_t)b * C * N;
  v8f acc[4] = {};

  for (int k = 0; k < C; k += 4) {
    float4 a4 = *(const float4*)(arow + k);
    v2f a = selpair(a4, hw);
    const float* bp = yb + (size_t)(k + 2 * hw) * N + m0 + l15;
#pragma unroll
    for (int t = 0; t < 4; ++t) {
      v2f bv;
      bv.x = bp[16 * t];
      bv.y = bp[N + 16 * t];
      acc[t] = WMMA_F32(acc[t], a, bv);
    }
  }

  const float gm = gamma[0];
  const float* xb = x + (size_t)b * C * N;
  float* ob = out + (size_t)b * C * N;
#pragma unroll
  for (int r = 0; r < 8; ++r) {
    const int row = o0 + r + 8 * hw;
#pragma unroll
    for (int t = 0; t < 4; ++t) {
      const size_t idx = (size_t)row * N + m0 + 16 * t + l15;
      ob[idx] = fmaf(acc[t][r], gm, xb[idx]);
    }
  }
}

// ---------------------------------------------------------------------------
extern "C" void kernel_launch(void* const* d_in, const int* in_sizes, int n_in,
                              void* d_out, int out_size, void* d_ws, size_t ws_size,
                              hipStream_t stream) {
  (void)in_sizes; (void)n_in; (void)out_size; (void)ws_size;
  const float* x       = (const float*)d_in[0];
  const float* w_phi   = (const float*)d_in[1];
  const float* w_theta = (const float*)d_in[2];
  const float* w_g     = (const float*)d_in[3];
  const float* w_mask  = (const float*)d_in[4];
  const float* gamma   = (const float*)d_in[5];
  float* out = (float*)d_out;
  float* ws  = (float*)d_ws;

  const size_t CN = (size_t)C * N;  // per-batch [C,N] = 2M floats
  float* phi   = ws;            // 8*CN
  float* theta = ws + 8 * CN;   // 8*CN
  float* g     = ws + 16 * CN;  // 8*CN
  float* attn  = ws + 24 * CN;  // B*C*C = 2M floats
  float* yt    = ws;            // reuse phi region (dead after cam_logits)

  dim3 blk(128);
  cam_proj  <<<dim3(64, 8, 24), blk, 0, stream>>>(x, w_phi, w_theta, w_g, phi, theta, g);
  cam_logits<<<dim3(8, 8, 8),   blk, 0, stream>>>(phi, theta, attn);
  cam_softmax<<<dim3(4096), dim3(128), 0, stream>>>(attn);
  cam_apply <<<dim3(64, 8, 8),  blk, 0, stream>>>(attn, g, yt);
  cam_mask  <<<dim3(64, 8, 8),  blk, 0, stream>>>(yt, w_mask, x, gamma, out);
}